// LINEAR_JIT_27633819583094
// MI455X (gfx1250) — compile-verified
//
#include <hip/hip_runtime.h>

// Problem dimensions (fixed by the reference).
#define NB 128     // batch
#define TT 2048    // timesteps
#define II 128     // input dim
#define HH 1024    // hidden dim

typedef float v2f __attribute__((ext_vector_type(2)));
typedef float v8f __attribute__((ext_vector_type(8)));

// One wave (32 threads / block) computes a 16(M) x 64(N) tile of
//   C = A * W^T (+ bias[n]) (+ addRow[m,n])
// A: M x K row-major, row stride lda
// W: Nout x K row-major (so C[m][n] = sum_k A[m][k] * W[n][k])
// Four independent WMMA accumulator chains (one per 16-column group) give ILP
// on the serial K reduction; the A fragment is loaded once per K-chunk and
// reused by all four V_WMMA_F32_16X16X4_F32 issues.
__device__ __forceinline__ void gemm_tile_16x64(
    const float* __restrict__ A, size_t lda,
    const float* __restrict__ W, int K,
    float* __restrict__ C, size_t ldc,
    const float* __restrict__ addRow, size_t ldadd,   // optional per-element addend
    const float* __restrict__ bias)                   // optional per-column bias
{
    const int lane = (int)(threadIdx.x & 31);
    const int half = lane >> 4;      // 0: lanes 0-15, 1: lanes 16-31
    const int l16  = lane & 15;

    const size_t m0 = (size_t)blockIdx.y * 16;
    const size_t n0 = (size_t)blockIdx.x * 64;

    v8f acc0 = {}, acc1 = {}, acc2 = {}, acc3 = {};

    // Per-lane base pointers. A slot (half, v) carries K = 2*half + v; B uses
    // the identical convention, so the pairing inside the MAC unit matches.
    const float* aPtr  = A + (m0 + (size_t)l16) * lda + 2 * half;
    const float* bPtr0 = W + (n0 + 0 * 16 + (size_t)l16) * (size_t)K + 2 * half;
    const float* bPtr1 = W + (n0 + 1 * 16 + (size_t)l16) * (size_t)K + 2 * half;
    const float* bPtr2 = W + (n0 + 2 * 16 + (size_t)l16) * (size_t)K + 2 * half;
    const float* bPtr3 = W + (n0 + 3 * 16 + (size_t)l16) * (size_t)K + 2 * half;

    for (int k = 0; k < K; k += 4) {
        v2f a  = *(const v2f*)(aPtr  + k);
        v2f b0 = *(const v2f*)(bPtr0 + k);
        v2f b1 = *(const v2f*)(bPtr1 + k);
        v2f b2 = *(const v2f*)(bPtr2 + k);
        v2f b3 = *(const v2f*)(bPtr3 + k);
        acc0 = __builtin_amdgcn_wmma_f32_16x16x4_f32(false, a, false, b0,
                                                     (short)0, acc0, false, false);
        acc1 = __builtin_amdgcn_wmma_f32_16x16x4_f32(false, a, false, b1,
                                                     (short)0, acc1, false, false);
        acc2 = __builtin_amdgcn_wmma_f32_16x16x4_f32(false, a, false, b2,
                                                     (short)0, acc2, false, false);
        acc3 = __builtin_amdgcn_wmma_f32_16x16x4_f32(false, a, false, b3,
                                                     (short)0, acc3, false, false);
    }

    // Epilogue: documented 16x16 f32 C/D layout:
    //   VGPR r, lanes 0-15 -> M = r,     N = lane
    //   VGPR r, lanes16-31 -> M = r + 8, N = lane - 16
    v8f accs[4] = {acc0, acc1, acc2, acc3};
#pragma unroll
    for (int j = 0; j < 4; ++j) {
        const size_t n = n0 + (size_t)(j * 16 + l16);
        const float badd = bias ? bias[n] : 0.0f;
#pragma unroll
        for (int r = 0; r < 8; ++r) {
            const size_t m = m0 + (size_t)(r + 8 * half);
            float v = accs[j][r] + badd;
            if (addRow) v += addRow[m * ldadd + n];
            C[m * ldc + n] = v;
        }
    }
}

// proj[m, n] = sum_i x[m, i] * w_ih[n, i] + b_ih[n];  m in [0, N*T), contiguous.
__global__ void proj_kernel(const float* __restrict__ x,
                            const float* __restrict__ w_ih,
                            const float* __restrict__ b_ih,
                            float* __restrict__ proj)
{
    gemm_tile_16x64(x, II, w_ih, II, proj, HH, nullptr, 0, b_ih);
}

// hidden[:, 0, :] = initial + proj[:, 0, :]
__global__ void h0_kernel(const float* __restrict__ initial,
                          const float* __restrict__ proj,
                          float* __restrict__ hidden)
{
    const size_t i = (size_t)blockIdx.x * blockDim.x + threadIdx.x;  // over N*H
    const size_t n = i / HH;
    const size_t h = i % HH;
    const size_t off = n * (size_t)TT * HH + h;
    hidden[off] = initial[i] + proj[off];
}

// hidden[:, t, :] = hidden[:, t-1, :] @ w_hh^T + proj[:, t-1, :]
// Pointers come in pre-offset to timestep; row stride between batch rows is T*H.
__global__ void step_kernel(const float* __restrict__ h_prev,
                            const float* __restrict__ w_hh,
                            const float* __restrict__ p_prev,
                            float* __restrict__ h_out)
{
    const size_t strideNTH = (size_t)TT * HH;
    gemm_tile_16x64(h_prev, strideNTH, w_hh, HH, h_out, strideNTH,
                    p_prev, strideNTH, nullptr);
}

extern "C" void kernel_launch(void* const* d_in, const int* in_sizes, int n_in,
                              void* d_out, int out_size, void* d_ws, size_t ws_size,
                              hipStream_t stream)
{
    (void)in_sizes; (void)n_in; (void)out_size; (void)d_ws; (void)ws_size;

    const float* x       = (const float*)d_in[0];  // [N, T, I]
    const float* w_ih    = (const float*)d_in[1];  // [H, I]
    const float* b_ih    = (const float*)d_in[2];  // [H]
    const float* w_hh    = (const float*)d_in[3];  // [H, H]
    const float* initial = (const float*)d_in[4];  // [N, H]

    float* hidden = (float*)d_out;                       // output copy 1: [N, T, H]
    float* proj   = hidden + (size_t)NB * TT * HH;       // output copy 2 region = proj scratch

    const dim3 waveBlk(32);

    // Phase 1: input projection GEMM (262144 x 128 x 1024), fully parallel.
    {
        dim3 grid(HH / 64, (NB * TT) / 16);  // (16, 16384)
        proj_kernel<<<grid, waveBlk, 0, stream>>>(x, w_ih, b_ih, proj);
    }

    // Phase 2: h0 = initial + proj[:, 0]
    {
        const int elems = NB * HH;  // 131072
        h0_kernel<<<elems / 256, 256, 0, stream>>>(initial, proj, hidden);
    }

    // Phase 3: sequential recurrence, one GEMM (128 x 1024 x 1024) per step.
    {
        dim3 grid(HH / 64, NB / 16);  // (16, 8) -> 128 waves per step
        for (int t = 1; t < TT; ++t) {
            step_kernel<<<grid, waveBlk, 0, stream>>>(
                hidden + (size_t)(t - 1) * HH,
                w_hh,
                proj + (size_t)(t - 1) * HH,
                hidden + (size_t)t * HH);
        }
    }

    // Phase 4: duplicate output (reference returns the tensor twice);
    // this overwrites the proj scratch with the final result.
    hipMemcpyAsync(proj, hidden, (size_t)NB * TT * HH * sizeof(float),
                   hipMemcpyDeviceToDevice, stream);
}